// AuxiliaryTaskPair_78606491452413
// MI455X (gfx1250) — compile-verified
//
#include <hip/hip_runtime.h>

// Problem constants (match reference)
#define B_  32
#define S_  2048
#define H_  1024
#define P_  16
#define L_  2
#define NPAIR (B_ * P_)      // 512
#define TWOH  (2 * H_)       // 2048

typedef float v2f __attribute__((ext_vector_type(2)));
typedef float v8f __attribute__((ext_vector_type(8)));

// ---------------------------------------------------------------------------
// Zero-fill the B-matrix padding row in workspace (TWOH floats = 8 KB).
// Lanes n >= L of the WMMA B-fragment read from this instead of being
// exec-predicated, keeping the GEMM inner loop branchless.
// ---------------------------------------------------------------------------
__global__ __launch_bounds__(256)
void zero_pad_kernel(float* __restrict__ zpad)
{
    const int i = blockIdx.x * 256 + threadIdx.x;
    if (i < TWOH) zpad[i] = 0.0f;
}

// ---------------------------------------------------------------------------
// Phase 1: span-mean gather.  One block (8 wave32s) per (b,p) pair.
// Thread t owns columns [4t, 4t+4): 128-bit loads, fully coalesced per row.
// Writes feats[pair, 0:1024] = mean(span1 rows), feats[pair, 1024:2048] = mean(span2).
// Also emits labels (as float) for the tuple output.
// ---------------------------------------------------------------------------
__global__ __launch_bounds__(256)
void span_mean_kernel(const float* __restrict__ seq,      // [B, S, H]
                      const int*   __restrict__ pidx,     // [B, P, 5]
                      float*       __restrict__ feats,    // [NPAIR, 2H] (workspace)
                      float*       __restrict__ labels)   // [NPAIR]
{
    const int pair = blockIdx.x;            // 0..511
    const int b    = pair / P_;
    const int tid  = threadIdx.x;           // 0..255
    const int col  = tid * 4;

    const int* pi = pidx + pair * 5;
    const int s1 = pi[0], e1 = pi[1], s2 = pi[2], e2 = pi[3], lab = pi[4];
    if (tid == 0) labels[pair] = (float)lab;

    const float* __restrict__ seqb = seq + (size_t)b * S_ * H_;
    float* __restrict__ frow = feats + (size_t)pair * TWOH;

    // ---- span 1 ----
    {
        float ax = 0.f, ay = 0.f, az = 0.f, aw = 0.f;
        #pragma unroll 2
        for (int t = s1; t < e1; ++t) {
            const float4 v = *(const float4*)(seqb + (size_t)t * H_ + col);
            // speculative prefetch a few rows ahead (global_prefetch_b8)
            __builtin_prefetch(seqb + (size_t)(t + 4) * H_ + col, 0, 0);
            ax += v.x; ay += v.y; az += v.z; aw += v.w;
        }
        const float inv = 1.0f / (float)(e1 - s1);
        float4 r; r.x = ax * inv; r.y = ay * inv; r.z = az * inv; r.w = aw * inv;
        *(float4*)(frow + col) = r;
    }
    // ---- span 2 ----
    {
        float ax = 0.f, ay = 0.f, az = 0.f, aw = 0.f;
        #pragma unroll 2
        for (int t = s2; t < e2; ++t) {
            const float4 v = *(const float4*)(seqb + (size_t)t * H_ + col);
            __builtin_prefetch(seqb + (size_t)(t + 4) * H_ + col, 0, 0);
            ax += v.x; ay += v.y; az += v.z; aw += v.w;
        }
        const float inv = 1.0f / (float)(e2 - s2);
        float4 r; r.x = ax * inv; r.y = ay * inv; r.z = az * inv; r.w = aw * inv;
        *(float4*)(frow + H_ + col) = r;
    }
}

// ---------------------------------------------------------------------------
// Phase 2: logits[512, 2] = feats[512, 2048] @ W^T[2048, 2] + b
// via V_WMMA_F32_16X16X4_F32.  One wave per 16-row tile (grid = 32 blocks,
// blockDim = 32).  N padded to 16: lanes with n >= L read B from a zero row,
// so the inner loop is branchless: 2x global_load_b64 + 1x v_wmma per step.
//
// ISA VGPR layouts (CDNA5 ISA 7.12.2), wave32:
//   A (16x4 f32, 2 VGPRs): lane l -> row m = l&15; VGPR j holds K = j + 2*(l>>4)
//   B (4x16 f32, 2 VGPRs): lane l -> col n = l&15; VGPR j holds K = j + 2*(l>>4)
//   C/D (16x16 f32, 8 VGPRs): lane l -> col n = l&15; VGPR j holds M = j + 8*(l>>4)
// ---------------------------------------------------------------------------
__global__ __launch_bounds__(32)
void proj_wmma_kernel(const float* __restrict__ feats,   // [NPAIR, 2H]
                      const float* __restrict__ W,       // [L, 2H]
                      const float* __restrict__ zpad,    // [2H] zeros
                      const float* __restrict__ bias,    // [L]
                      float*       __restrict__ logits)  // [NPAIR, L]
{
    const int tile = blockIdx.x;        // 16 rows per tile
    const int lane = threadIdx.x;       // 0..31
    const int m    = lane & 15;         // A row for this lane
    const int kh   = lane >> 4;         // K half select (0 or 1)
    const int n    = lane & 15;         // B/C column for this lane
    const bool nvalid = (n < L_);

    const float* __restrict__ arow = feats + (size_t)(tile * 16 + m) * TWOH;
    // Branchless B source: valid lanes read W row n, pad lanes read zeros.
    const float* __restrict__ wrow = nvalid ? (W + (size_t)n * TWOH) : zpad;

    v8f c = {0.f, 0.f, 0.f, 0.f, 0.f, 0.f, 0.f, 0.f};

    #pragma unroll 8
    for (int k0 = 0; k0 < TWOH; k0 += 4) {
        const int k = k0 + 2 * kh;
        v2f a  = *(const v2f*)(arow + k);   // global_load_b64
        v2f bm = *(const v2f*)(wrow + k);   // global_load_b64 (zeros for n>=L)
        // D = A(16x4) * B(4x16) + C   -> v_wmma_f32_16x16x4_f32
        c = __builtin_amdgcn_wmma_f32_16x16x4_f32(
                /*neg_a=*/false, a,
                /*neg_b=*/false, bm,
                /*c_mod=*/(short)0, c,
                /*reuse_a=*/false, /*reuse_b=*/false);
    }

    if (nvalid) {
        const float bv = bias[n];
        #pragma unroll
        for (int j = 0; j < 8; ++j) {
            const int mg = tile * 16 + j + 8 * kh;   // D row held by this lane/VGPR
            logits[(size_t)mg * L_ + n] = c[j] + bv;
        }
    }
}

// ---------------------------------------------------------------------------
// Launch
// ---------------------------------------------------------------------------
extern "C" void kernel_launch(void* const* d_in, const int* in_sizes, int n_in,
                              void* d_out, int out_size, void* d_ws, size_t ws_size,
                              hipStream_t stream) {
    const float* seq  = (const float*)d_in[0];   // [B, S, H] f32
    const int*   pidx = (const int*)  d_in[1];   // [B, P, 5] i32
    const float* W    = (const float*)d_in[2];   // [L, 2H]  f32
    const float* bias = (const float*)d_in[3];   // [L]      f32

    float* out    = (float*)d_out;
    float* logits = out;                 // NPAIR * L
    float* labels = out + NPAIR * L_;    // NPAIR (as float)

    float* feats = (float*)d_ws;                 // NPAIR * 2H floats = 4 MB
    float* zpad  = feats + (size_t)NPAIR * TWOH; // 2H floats = 8 KB zero row

    zero_pad_kernel<<<(TWOH + 255) / 256, 256, 0, stream>>>(zpad);
    span_mean_kernel<<<NPAIR, 256, 0, stream>>>(seq, pidx, feats, labels);
    proj_wmma_kernel<<<NPAIR / 16, 32, 0, stream>>>(feats, W, zpad, bias, logits);
}